// TRCAlayer_60112362275390
// MI455X (gfx1250) — compile-verified
//
#include <hip/hip_runtime.h>

typedef float v2f __attribute__((ext_vector_type(2)));
typedef float v8f __attribute__((ext_vector_type(8)));

#define B_   256
#define NM_  5
#define NC_  64
#define T_   1024
#define NF_  40

#define WAVES_PER_BLOCK 8
#define THREADS (WAVES_PER_BLOCK * 32)          // 256 threads = 8 wave32
#define TG (T_ / (16 * WAVES_PER_BLOCK))        // 8 t-tile groups per (b,m)
#define LDS_STRIDE 68                            // padded W row (floats), bank-conflict-free b64 reads
#define MT 3                                     // M tiles: 40 rows padded to 48

__global__ __launch_bounds__(THREADS)
void trca_wmma_kernel(const float* __restrict__ x,
                      const float* __restrict__ W,
                      float* __restrict__ out)
{
    __shared__ float Wlds[48 * LDS_STRIDE];      // 13,056 B

    const int bid    = blockIdx.x;
    const int tgroup = bid % TG;
    const int bm     = bid / TG;                 // 0 .. B*NM-1
    const int m      = bm % NM_;

    // ---- stage W[m] (40x64, zero-padded to 48 rows) into LDS ----
    const float* Wm = W + m * (NF_ * NC_);
    for (int i = threadIdx.x; i < 48 * NC_; i += THREADS) {
        const int f = i >> 6;
        const int c = i & 63;
        Wlds[f * LDS_STRIDE + c] = (f < NF_) ? Wm[f * NC_ + c] : 0.0f;
    }
    __syncthreads();

    const int wave   = threadIdx.x >> 5;
    const int lane   = threadIdx.x & 31;
    const int lane16 = lane & 15;
    const int hi     = (lane >= 16) ? 1 : 0;     // upper half-wave handles K=+2,+3 / M=+8

    const int tbase = (tgroup * WAVES_PER_BLOCK + wave) * 16;

    // Per-lane x base: x[b,m, c = 2*hi, t = tbase + lane16]
    const float* xb = x + (size_t)bm * (NC_ * T_) + (size_t)(2 * hi) * T_
                        + (size_t)(tbase + lane16);

    v8f acc[MT] = {};                             // 24 accumulator VGPRs

    #pragma unroll
    for (int k = 0; k < NC_ / 4; ++k) {
        // B operand (4x16 K-by-N tile of x): rows c0, c0+1 at this lane's t.
        // c0 = 4k for lanes 0-15, 4k+2 for lanes 16-31 (ISA B-layout).
        v2f bv;
        bv.x = xb[k * 4 * T_];                    // immediate-offset global_load_b32
        bv.y = xb[k * 4 * T_ + T_];

        const int coff = k * 4 + 2 * hi;
        #pragma unroll
        for (int mt = 0; mt < MT; ++mt) {
            // A operand (16x4 M-by-K tile of W): ds_load_b64, conflict-free via stride 68
            v2f av = *(const v2f*)&Wlds[(mt * 16 + lane16) * LDS_STRIDE + coff];
            acc[mt] = __builtin_amdgcn_wmma_f32_16x16x4_f32(
                /*neg_a=*/false, av, /*neg_b=*/false, bv,
                /*c_mod=*/(short)0, acc[mt],
                /*reuse_a=*/false, /*reuse_b=*/false);
        }
    }

    // ---- store: out[b,m,f,t]; C/D layout: VGPR r, lanes 0-15 -> M=r, lanes 16-31 -> M=r+8
    float* ob = out + (size_t)bm * (NF_ * T_) + (size_t)(tbase + lane16);
    const int rowoff = 8 * hi;
    #pragma unroll
    for (int mt = 0; mt < MT; ++mt) {
        #pragma unroll
        for (int r = 0; r < 8; ++r) {
            const int f = mt * 16 + r + rowoff;
            if (f < NF_) ob[(size_t)f * T_] = acc[mt][r];
        }
    }
}

extern "C" void kernel_launch(void* const* d_in, const int* in_sizes, int n_in,
                              void* d_out, int out_size, void* d_ws, size_t ws_size,
                              hipStream_t stream) {
    const float* x   = (const float*)d_in[0];    // (256, 5, 64, 1024) fp32
    const float* W   = (const float*)d_in[1];    // (5, 40, 64) fp32
    float*       out = (float*)d_out;            // (256, 5, 40, 1024) fp32

    dim3 grid(B_ * NM_ * TG);                    // 10240 blocks
    dim3 block(THREADS);                         // 256 threads (8 wave32)
    hipLaunchKernelGGL(trca_wmma_kernel, grid, block, 0, stream, x, W, out);
}